// DeformableTransformerEncoder_67826123538674
// MI455X (gfx1250) — compile-verified
//
#include <hip/hip_runtime.h>
#include <math.h>

// ---------------------------------------------------------------------------
// DeformableTransformerEncoder for MI455X (gfx1250, wave32, WMMA).
// All GEMMs run on the matrix pipe via V_WMMA_F32_16X16X4_F32 (full f32 —
// preserves reference precision; total ~180 GFLOP, working set is L2-resident,
// so a clean f32 matrix-pipe loop is the right operating point).
// Workspace requirement: ~330 MB of f32 scratch.
// ---------------------------------------------------------------------------

#define D_MODEL   256
#define NTOK      17821
#define KSEL      8911     // int(NTOK*0.5)+1
#define BATCH     2
#define LN_EPS    1e-5f
#define NUM_LAYERS 6

typedef float v2f __attribute__((ext_vector_type(2)));
typedef float v8f __attribute__((ext_vector_type(8)));

__device__ __forceinline__ float apply_act(float x, int act)
{
    if (act == 1) return 0.5f * x * (1.0f + erff(x * 0.70710678118654752f));
    if (act == 2) return fmaxf(x, 0.0f);
    return x;
}

// ---------------------------------------------------------------------------
// f32 WMMA GEMM:  C[M,N] = act(A[M,K] @ B[K,N] + bias[N])
// Block = 128 threads (4 waves). Each wave owns a 32x32 output tile
// (2x2 WMMA fragments): per K-step of 4 -> 2 A b64 loads, 4 B b32 loads,
// 4 v_wmma_f32_16x16x4_f32. Out-of-range A rows are clamped (not masked):
// they only pollute accumulator rows the epilogue never stores, so the loop
// is branch-free and EXEC stays all-1s at every WMMA (ISA 7.12 requirement).
// Fragment layouts per ISA 7.12.2:
//   A 16x4 : lane l -> row m=l&15, VGPR{0,1} = K + 2*(l>>4) + {0,1}
//   B 4x16 : lane l -> col n=l&15, same K split
//   C 16x16: vgpr v, lane l -> m = v + 8*(l>>4), n = l&15
// ---------------------------------------------------------------------------
__global__ void gemm_f32_wmma(const float* __restrict__ A, const float* __restrict__ B,
                              const float* __restrict__ bias, float* __restrict__ C,
                              int M, int K, int N, int act)
{
    const int lane = threadIdx.x & 31;
    const int wid  = threadIdx.x >> 5;
    const int m0   = blockIdx.x * 32;
    const int n0   = (blockIdx.y * 4 + wid) * 32;
    if (n0 >= N) return;                       // uniform per wave

    const int half = lane >> 4;                // 0 or 1
    const int l15  = lane & 15;
    const int koff = half * 2;

    // A row pointers for the two M sub-tiles (clamped; see header comment)
    const int am0 = m0 + l15;
    const int am1 = m0 + 16 + l15;
    const float* a0p = A + (size_t)(am0 < M ? am0 : M - 1) * K + koff;
    const float* a1p = A + (size_t)(am1 < M ? am1 : M - 1) * K + koff;

    // B column pointers for the two N sub-tiles (N is a multiple of 32 here)
    const int cn0 = n0 + l15;
    const int cn1 = n0 + 16 + l15;
    const float* b0p = B + (size_t)koff * N + cn0;
    const float* b1p = B + (size_t)koff * N + cn1;
    const size_t rowstep = (size_t)4 * N;

    v8f acc00 = {}, acc01 = {}, acc10 = {}, acc11 = {};
#pragma unroll 2
    for (int k = 0; k < K; k += 4) {
        v2f a0 = *(const v2f*)a0p; a0p += 4;   // 8B-aligned (K % 4 == 0)
        v2f a1 = *(const v2f*)a1p; a1p += 4;
        v2f b0, b1;
        b0.x = b0p[0]; b0.y = b0p[N];
        b1.x = b1p[0]; b1.y = b1p[N];
        b0p += rowstep; b1p += rowstep;
        acc00 = __builtin_amdgcn_wmma_f32_16x16x4_f32(false, a0, false, b0, (short)0, acc00, false, false);
        acc01 = __builtin_amdgcn_wmma_f32_16x16x4_f32(false, a0, false, b1, (short)0, acc01, false, false);
        acc10 = __builtin_amdgcn_wmma_f32_16x16x4_f32(false, a1, false, b0, (short)0, acc10, false, false);
        acc11 = __builtin_amdgcn_wmma_f32_16x16x4_f32(false, a1, false, b1, (short)0, acc11, false, false);
    }

    const float bv0 = bias ? bias[cn0] : 0.0f;
    const float bv1 = bias ? bias[cn1] : 0.0f;
#pragma unroll
    for (int v = 0; v < 8; ++v) {
        const int r  = v + half * 8;
        const int cmA = m0 + r;
        const int cmB = m0 + 16 + r;
        if (cmA < M) {
            C[(size_t)cmA * N + cn0] = apply_act(acc00[v] + bv0, act);
            C[(size_t)cmA * N + cn1] = apply_act(acc01[v] + bv1, act);
        }
        if (cmB < M) {
            C[(size_t)cmB * N + cn0] = apply_act(acc10[v] + bv0, act);
            C[(size_t)cmB * N + cn1] = apply_act(acc11[v] + bv1, act);
        }
    }
}

// ---------------------------------------------------------------------------
// LayerNorm over 256 channels, optional residual:  out = LN(x (+ r)) * g + b
// One wave per row (8 f32 per lane), wave reduction via shfl_xor.
// ---------------------------------------------------------------------------
__global__ void add_ln256(const float* __restrict__ x, const float* __restrict__ r,
                          const float* __restrict__ g, const float* __restrict__ bta,
                          float* __restrict__ out, int rows)
{
    const int wid  = threadIdx.x >> 5;
    const int lane = threadIdx.x & 31;
    const int row  = blockIdx.x * 8 + wid;
    if (row >= rows) return;
    const float* xr = x + (size_t)row * 256;
    const float* rr = r ? r + (size_t)row * 256 : nullptr;
    float v[8];
    float s = 0.0f, sq = 0.0f;
#pragma unroll
    for (int i = 0; i < 8; ++i) {
        float t = xr[lane + i * 32];
        if (rr) t += rr[lane + i * 32];
        v[i] = t; s += t; sq += t * t;
    }
#pragma unroll
    for (int d = 16; d >= 1; d >>= 1) {
        s  += __shfl_xor(s,  d, 32);
        sq += __shfl_xor(sq, d, 32);
    }
    const float mean = s * (1.0f / 256.0f);
    const float var  = sq * (1.0f / 256.0f) - mean * mean;
    const float inv  = rsqrtf(var + LN_EPS);
    float* orow = out + (size_t)row * 256;
#pragma unroll
    for (int i = 0; i < 8; ++i) {
        const int c = lane + i * 32;
        orow[c] = (v[i] - mean) * inv * g[c] + bta[c];
    }
}

// ---------------------------------------------------------------------------
// Mask-predictor helpers
// ---------------------------------------------------------------------------
__global__ void zero_k(float* p, int n)
{
    int i = blockIdx.x * blockDim.x + threadIdx.x;
    if (i < n) p[i] = 0.0f;
}

// Partial sums of z1[:, :, 128:256] over tokens -> gmean[b,128]
__global__ void gmean_partial(const float* __restrict__ z1, float* __restrict__ gmean)
{
    const int c = threadIdx.x;                 // 0..127
    const int b = blockIdx.y;
    const int per = (NTOK + gridDim.x - 1) / gridDim.x;
    const int i0 = blockIdx.x * per;
    const int i1 = (i0 + per < NTOK) ? i0 + per : NTOK;
    float s = 0.0f;
    for (int i = i0; i < i1; ++i)
        s += z1[((size_t)b * NTOK + i) * 256 + 128 + c];
    atomicAdd(&gmean[b * 128 + c], s);
}

// Overwrite z1[:, :, 128:256] with broadcast mean
__global__ void gmean_broadcast(float* __restrict__ z1, const float* __restrict__ gmean)
{
    size_t t = (size_t)blockIdx.x * blockDim.x + threadIdx.x;   // B*NTOK*128
    if (t >= (size_t)BATCH * NTOK * 128) return;
    const int c = (int)(t & 127);
    const size_t tok = t >> 7;
    const int b = (int)(tok / NTOK);
    z1[tok * 256 + 128 + c] = gmean[b * 128 + c] * (1.0f / (float)NTOK);
}

// score = z3 @ w4 + b4 (64 -> 1)
__global__ void score_k(const float* __restrict__ z3, const float* __restrict__ w4,
                        const float* __restrict__ b4, float* __restrict__ score)
{
    int t = blockIdx.x * blockDim.x + threadIdx.x;
    if (t >= BATCH * NTOK) return;
    const float* zr = z3 + (size_t)t * 64;
    float s = b4[0];
#pragma unroll 8
    for (int j = 0; j < 64; ++j) s += zr[j] * w4[j];
    score[t] = s;
}

// Exact, stable top-K by rank counting (matches jax.lax.top_k ordering).
__global__ void topk_rank(const float* __restrict__ score, int* __restrict__ topk)
{
    int t = blockIdx.x * blockDim.x + threadIdx.x;
    if (t >= BATCH * NTOK) return;
    const int b = t / NTOK, i = t % NTOK;
    const float* sc = score + (size_t)b * NTOK;
    const float my = sc[i];
    int rank = 0;
    for (int j = 0; j < NTOK; ++j) {
        const float sj = sc[j];
        rank += (sj > my) || (sj == my && j < i);
    }
    if (rank < KSEL) topk[b * KSEL + rank] = i;
}

// Gather tgt/pos_k rows, compute reference points for the selected tokens.
__global__ void gather_k(const float* __restrict__ src, const float* __restrict__ pos,
                         const float* __restrict__ vr, const int* __restrict__ topk,
                         float* __restrict__ tgt, float* __restrict__ posk,
                         float* __restrict__ refk)
{
    const int bq = blockIdx.x;                 // 0 .. B*KSEL-1
    const int b  = bq / KSEL;
    const int i  = topk[bq];
    const int c  = threadIdx.x;                // 256
    tgt [(size_t)bq * 256 + c] = src[((size_t)b * NTOK + i) * 256 + c];
    posk[(size_t)bq * 256 + c] = pos[((size_t)b * NTOK + i) * 256 + c];
    if (c == 0) {
        const int Hs[4] = {100, 50, 25, 13};
        const int Ws[4] = {134, 67, 34, 17};
        const int St[4] = {0, 13400, 16750, 17600};
        int L = 3;
        for (int l = 0; l < 3; ++l) { if (i < St[l + 1]) { L = l; break; } }
        const int local = i - St[L];
        const int y = local / Ws[L], x = local % Ws[L];
        const float ry = (y + 0.5f) / (vr[(b * 4 + L) * 2 + 1] * (float)Hs[L]);
        const float rx = (x + 0.5f) / (vr[(b * 4 + L) * 2 + 0] * (float)Ws[L]);
        for (int l = 0; l < 4; ++l) {
            refk[((size_t)bq * 4 + l) * 2 + 0] = rx * vr[(b * 4 + l) * 2 + 0];
            refk[((size_t)bq * 4 + l) * 2 + 1] = ry * vr[(b * 4 + l) * 2 + 1];
        }
    }
}

__global__ void add_ew(const float* __restrict__ a, const float* __restrict__ b,
                       float* __restrict__ o, int n)
{
    int i = blockIdx.x * blockDim.x + threadIdx.x;
    if (i < n) o[i] = a[i] + b[i];
}

// Softmax over the 16 (level,point) attention logits per (b,q,head).
__global__ void softmax16(float* __restrict__ aw, int total)
{
    int t = blockIdx.x * blockDim.x + threadIdx.x;
    if (t >= total) return;
    float* p = aw + (size_t)(t >> 3) * 128 + (t & 7) * 16;
    float m = p[0];
#pragma unroll
    for (int j = 1; j < 16; ++j) m = fmaxf(m, p[j]);
    float e[16], s = 0.0f;
#pragma unroll
    for (int j = 0; j < 16; ++j) { e[j] = expf(p[j] - m); s += e[j]; }
    const float inv = 1.0f / s;
#pragma unroll
    for (int j = 0; j < 16; ++j) p[j] = e[j] * inv;
}

// Multi-scale deformable bilinear sampling + attention-weighted reduce.
// One wave per (b, q, head); 32 lanes = D_HEAD channels.
__global__ void deform_sample(const float* __restrict__ value, const float* __restrict__ offb,
                              const float* __restrict__ awb, const float* __restrict__ refk,
                              float* __restrict__ attn)
{
    const int wid  = threadIdx.x >> 5;
    const int lane = threadIdx.x & 31;
    const int g = blockIdx.x * 8 + wid;                 // over B*KSEL*8
    if (g >= BATCH * KSEL * 8) return;
    const int h  = g & 7;
    const int bq = g >> 3;
    const int b  = bq / KSEL;
    const int Hs[4] = {100, 50, 25, 13};
    const int Ws[4] = {134, 67, 34, 17};
    const int St[4] = {0, 13400, 16750, 17600};

    const float* offr = offb + (size_t)bq * 256;
    const float* awr  = awb  + (size_t)bq * 128 + h * 16;
    float acc = 0.0f;
#pragma unroll
    for (int l = 0; l < 4; ++l) {
        const int Hi = Hs[l], Wi = Ws[l];
        const float Hf = (float)Hi, Wf = (float)Wi;
        const float refx = refk[((size_t)bq * 4 + l) * 2 + 0];
        const float refy = refk[((size_t)bq * 4 + l) * 2 + 1];
        const float* vbase = value + (((size_t)b * NTOK + St[l]) * 8 + h) * 32 + lane;
#pragma unroll
        for (int p = 0; p < 4; ++p) {
            const float ox = offr[((h * 4 + l) * 4 + p) * 2 + 0];
            const float oy = offr[((h * 4 + l) * 4 + p) * 2 + 1];
            const float w  = awr[l * 4 + p];
            const float xf = (refx + ox / Wf) * Wf - 0.5f;
            const float yf = (refy + oy / Hf) * Hf - 0.5f;
            const float x0f = floorf(xf), y0f = floorf(yf);
            const float lx = xf - x0f, ly = yf - y0f;
            const int x0 = (int)x0f, y0 = (int)y0f;
            float s = 0.0f;
#pragma unroll
            for (int cy = 0; cy < 2; ++cy) {
#pragma unroll
                for (int cx = 0; cx < 2; ++cx) {
                    const int yi = y0 + cy, xi = x0 + cx;
                    const bool ok = (xi >= 0) && (xi < Wi) && (yi >= 0) && (yi < Hi);
                    const int yc = yi < 0 ? 0 : (yi > Hi - 1 ? Hi - 1 : yi);
                    const int xc = xi < 0 ? 0 : (xi > Wi - 1 ? Wi - 1 : xi);
                    float gv = vbase[(size_t)(yc * Wi + xc) * 256];   // token stride = 8*32
                    gv = ok ? gv : 0.0f;
                    const float ww = (cx ? lx : 1.0f - lx) * (cy ? ly : 1.0f - ly);
                    s += gv * ww;
                }
            }
            acc += w * s;
        }
    }
    attn[(size_t)bq * 256 + h * 32 + lane] = acc;
}

// Scatter tgt rows back into output at the top-k positions.
__global__ void scatter_k(const float* __restrict__ tgt, const int* __restrict__ topk,
                          float* __restrict__ out)
{
    const int bq = blockIdx.x;
    const int b  = bq / KSEL;
    const int i  = topk[bq];
    const int c  = threadIdx.x;
    out[((size_t)b * NTOK + i) * 256 + c] = tgt[(size_t)bq * 256 + c];
}

// ---------------------------------------------------------------------------
static inline void launch_gemm(const float* A, const float* B, const float* bias,
                               float* C, int M, int K, int N, int act, hipStream_t s)
{
    dim3 grid((M + 31) / 32, (N + 127) / 128);
    gemm_f32_wmma<<<grid, 128, 0, s>>>(A, B, bias, C, M, K, N, act);
}

extern "C" void kernel_launch(void* const* d_in, const int* in_sizes, int n_in,
                              void* d_out, int out_size, void* d_ws, size_t ws_size,
                              hipStream_t stream)
{
    (void)in_sizes; (void)n_in; (void)out_size; (void)ws_size;
    // setup_inputs() insertion order: src, spatial_shapes, level_start_index,
    // valid_ratios, pos, padding_mask, params{mp{...}, layer{...}} (leaves in
    // insertion order). spatial_shapes / level_start_index / padding_mask are
    // compile-time constants or all-false in this problem and are not read.
    const float* src = (const float*)d_in[0];
    const float* vr  = (const float*)d_in[3];
    const float* pos = (const float*)d_in[4];
    const float* ln_s = (const float*)d_in[6];
    const float* ln_b = (const float*)d_in[7];
    const float* w1 = (const float*)d_in[8];   const float* b1 = (const float*)d_in[9];
    const float* w2 = (const float*)d_in[10];  const float* b2 = (const float*)d_in[11];
    const float* w3 = (const float*)d_in[12];  const float* b3 = (const float*)d_in[13];
    const float* w4 = (const float*)d_in[14];  const float* b4 = (const float*)d_in[15];
    const float* val_w = (const float*)d_in[16]; const float* val_b = (const float*)d_in[17];
    const float* off_w = (const float*)d_in[18]; const float* off_b = (const float*)d_in[19];
    const float* aw_w  = (const float*)d_in[20]; const float* aw_b  = (const float*)d_in[21];
    const float* out_w = (const float*)d_in[22]; const float* out_b = (const float*)d_in[23];
    const float* n1_s  = (const float*)d_in[24]; const float* n1_b  = (const float*)d_in[25];
    const float* f1_w  = (const float*)d_in[26]; const float* f1_b  = (const float*)d_in[27];
    const float* f2_w  = (const float*)d_in[28]; const float* f2_b  = (const float*)d_in[29];
    const float* n2_s  = (const float*)d_in[30]; const float* n2_b  = (const float*)d_in[31];

    float* out = (float*)d_out;

    // ---- workspace carve (f32 elements) ----
    float* w = (float*)d_ws;
    size_t o = 0;
    const size_t BT = (size_t)BATCH * NTOK;     // 35642
    const size_t BK = (size_t)BATCH * KSEL;     // 17822
    float* xln   = w + o; o += BT * 256;
    float* z1    = w + o; o += BT * 256;
    float* z2    = w + o; o += BT * 128;
    float* z3    = w + o; o += BT * 64;
    float* score = w + o; o += BT;
    float* gmean = w + o; o += (size_t)BATCH * 128;
    int*   topk  = (int*)(w + o); o += BK;
    float* tgt   = w + o; o += BK * 256;
    float* posk  = w + o; o += BK * 256;
    float* refk  = w + o; o += BK * 8;
    float* qb    = w + o; o += BK * 256;
    float* valb  = w + o; o += BT * 256;
    float* offbf = w + o; o += BK * 256;
    float* awbf  = w + o; o += BK * 128;
    float* attn  = w + o; o += BK * 256;
    float* t2    = w + o; o += BK * 256;
    float* ffn   = w + o; o += BK * 1024;

    // ---- output starts as a copy of src ----
    hipMemcpyAsync(out, src, BT * 256 * sizeof(float), hipMemcpyDeviceToDevice, stream);

    // ---- mask predictor ----
    add_ln256<<<dim3((unsigned)((BT + 7) / 8)), 256, 0, stream>>>(src, nullptr, ln_s, ln_b, xln, (int)BT);
    launch_gemm(xln, w1, b1, z1, (int)BT, 256, 256, /*gelu*/1, stream);
    zero_k<<<1, 256, 0, stream>>>(gmean, BATCH * 128);
    gmean_partial<<<dim3(64, BATCH), 128, 0, stream>>>(z1, gmean);
    gmean_broadcast<<<dim3((unsigned)((BT * 128 + 255) / 256)), 256, 0, stream>>>(z1, gmean);
    launch_gemm(z1, w2, b2, z2, (int)BT, 256, 128, 1, stream);
    launch_gemm(z2, w3, b3, z3, (int)BT, 128, 64, 1, stream);
    score_k<<<dim3((unsigned)((BT + 255) / 256)), 256, 0, stream>>>(z3, w4, b4, score);

    // ---- top-k + gather ----
    topk_rank<<<dim3((unsigned)((BT + 255) / 256)), 256, 0, stream>>>(score, topk);
    gather_k<<<dim3((unsigned)BK), 256, 0, stream>>>(src, pos, vr, topk, tgt, posk, refk);

    // ---- encoder layers ----
    for (int layer = 0; layer < NUM_LAYERS; ++layer) {
        // value = output @ val_w + val_b  (over all N tokens)
        launch_gemm(out, val_w, val_b, valb, (int)BT, 256, 256, 0, stream);
        // q = tgt + pos_k
        add_ew<<<dim3((unsigned)((BK * 256 + 255) / 256)), 256, 0, stream>>>(tgt, posk, qb, (int)(BK * 256));
        // sampling offsets & attention weights
        launch_gemm(qb, off_w, off_b, offbf, (int)BK, 256, 256, 0, stream);
        launch_gemm(qb, aw_w, aw_b, awbf, (int)BK, 256, 128, 0, stream);
        softmax16<<<dim3((unsigned)((BK * 8 + 255) / 256)), 256, 0, stream>>>(awbf, (int)(BK * 8));
        // bilinear sampling + weighted reduce
        deform_sample<<<dim3((unsigned)((BK * 8 + 7) / 8)), 256, 0, stream>>>(valb, offbf, awbf, refk, attn);
        // output projection, residual + LN1
        launch_gemm(attn, out_w, out_b, t2, (int)BK, 256, 256, 0, stream);
        add_ln256<<<dim3((unsigned)((BK + 7) / 8)), 256, 0, stream>>>(tgt, t2, n1_s, n1_b, tgt, (int)BK);
        // FFN, residual + LN2
        launch_gemm(tgt, f1_w, f1_b, ffn, (int)BK, 256, 1024, /*relu*/2, stream);
        launch_gemm(ffn, f2_w, f2_b, t2, (int)BK, 1024, 256, 0, stream);
        add_ln256<<<dim3((unsigned)((BK + 7) / 8)), 256, 0, stream>>>(tgt, t2, n2_s, n2_b, tgt, (int)BK);
        // scatter updated queries into output
        scatter_k<<<dim3((unsigned)BK), 256, 0, stream>>>(tgt, topk, out);
    }
}